// MEMOIR_4922032521692
// MI455X (gfx1250) — compile-verified
//
#include <hip/hip_runtime.h>
#include <stdint.h>

typedef __attribute__((ext_vector_type(2))) float v2f;
typedef __attribute__((ext_vector_type(8))) float v8f;

#define D_DIM 4096
#define O_DIM 4096
#define S_DIM 2048
#define B_DIM 4
#define M_DIM (B_DIM * S_DIM)   // 8192
#define TOPK 2048
#define NMASKS 100

// ---------------- Kernel 1: prompt mean over x[0, :pb+1, :] -> f[D] ----------------
__global__ void k_prompt_mean(const float* __restrict__ x, const int* __restrict__ pbp,
                              float* __restrict__ f) {
    int d = blockIdx.x * blockDim.x + threadIdx.x;
    int cnt = *pbp + 1;                       // 1025
    float s = 0.f;
    for (int si = 0; si < cnt; ++si) s += x[(size_t)si * D_DIM + d];
    f[d] = s / (float)cnt;
}

// ---------------- Kernel 2: exact top-k rank + permutation scatter -> mask bits ----
__global__ void k_topk_mask(const float* __restrict__ f, const int* __restrict__ perm,
                            unsigned* __restrict__ mbits_out) {
    __shared__ float av[D_DIM];
    __shared__ unsigned mbits[D_DIM / 32];
    int tid = threadIdx.x;
    for (int i = tid; i < D_DIM; i += 256) av[i] = fabsf(f[i]);
    for (int i = tid; i < D_DIM / 32; i += 256) mbits[i] = 0u;
    __syncthreads();
    for (int d = tid; d < D_DIM; d += 256) {
        float v = av[d];
        int cnt = 0;
        for (int j = 0; j < D_DIM; ++j) {
            float u = av[j];
            cnt += (int)((u > v) || (u == v && j < d));   // exact top_k tie order
        }
        if (cnt < TOPK) {
            int p = perm[d];
            atomicOr(&mbits[p >> 5], 1u << (p & 31));
        }
    }
    __syncthreads();
    for (int i = tid; i < D_DIM / 32; i += 256) mbits_out[i] = mbits[i];
}

// ---------------- Kernel 3: overlap popcount vs stored masks -----------------------
__global__ void k_overlap(const unsigned* __restrict__ mbits,
                          const unsigned char* __restrict__ stored,
                          int* __restrict__ ovcnt) {
    __shared__ int red[256];
    int n = blockIdx.x;
    int tid = threadIdx.x;
    const unsigned char* row = stored + (size_t)n * D_DIM;
    int c = 0;
    for (int d = tid; d < D_DIM; d += 256) {
        unsigned bit = (mbits[d >> 5] >> (d & 31)) & 1u;
        c += (int)(bit & (unsigned)(row[d] != 0));
    }
    red[tid] = c;
    __syncthreads();
    for (int s = 128; s > 0; s >>= 1) {
        if (tid < s) red[tid] += red[tid + s];
        __syncthreads();
    }
    if (tid == 0) ovcnt[n] = red[0];
}

// ---------------- Kernel 4: argmax-first + gate -> float final mask ----------------
__global__ void k_select(const int* __restrict__ ovcnt,
                         const unsigned char* __restrict__ stored,
                         float* __restrict__ fmask) {
    __shared__ int sel[2];
    if (threadIdx.x == 0) {
        int bi = 0, bc = ovcnt[0];
        for (int n = 1; n < NMASKS; ++n) {
            int c = ovcnt[n];
            if (c > bc) { bc = c; bi = n; }   // first-max semantics like jnp.argmax
        }
        float ov = (float)bc / (float)TOPK;
        sel[0] = bi;
        sel[1] = (ov >= 0.6f) ? 1 : 0;
    }
    __syncthreads();
    int bi = sel[0], g = sel[1];
    const unsigned char* row = stored + (size_t)bi * D_DIM;
    for (int d = threadIdx.x; d < D_DIM; d += 256)
        fmask[d] = (g && row[d]) ? 1.0f : 0.0f;
}

// ---------------- Kernel 5: fused single GEMM with V_WMMA_F32_16X16X4_F32 ----------
// out[m, o] = sum_d x[m, d] * (w[o, d] + fmask[d] * nw[o, d])
// 128x128 block tile, 8 waves (4x2), 32x64 per wave (2x4 WMMA tiles), BK=32.
// Register double buffering with W_eff fused at PREFETCH time, so only 8 float4
// (32 VGPRs) of prefetch state stay live across the WMMA phase -> no spills.
#define BM 128
#define BN 128
#define BK 32
#define LDSTRIDE (BK + 4)   // pad: 36 floats per row, keeps 16B alignment

__global__ __launch_bounds__(256, 1) void k_gemm_wmma(
    const float* __restrict__ x, const float* __restrict__ w,
    const float* __restrict__ nw, const float* __restrict__ fm,
    float* __restrict__ out) {
    __shared__ float As[BM][LDSTRIDE];
    __shared__ float Bs[BN][LDSTRIDE];

    const int tid  = threadIdx.x;
    const int lane = tid & 31;
    const int wave = tid >> 5;
    const int wm = wave >> 1;          // 0..3 (M direction)
    const int wn = wave & 1;           // 0..1 (N direction)
    const int bm = blockIdx.y * BM;
    const int bn = blockIdx.x * BN;

    const int hk   = lane >> 4;        // half-wave: selects K pair
    const int mrow = lane & 15;        // row/col within 16

    // Per-thread staging coordinates (4 float4 apiece for A and B tiles)
    int srow[4], sc4[4];
#pragma unroll
    for (int j = 0; j < 4; ++j) {
        int i   = tid + j * 256;       // 0..1023
        srow[j] = i >> 3;              // 0..127
        sc4[j]  = (i & 7) << 2;        // 0,4,...,28
    }

    v8f acc[2][4];
#pragma unroll
    for (int tm = 0; tm < 2; ++tm)
#pragma unroll
        for (int tn = 0; tn < 4; ++tn) acc[tm][tn] = (v8f){};

    // Only 32 VGPRs of prefetch state: A tile + already-fused B tile.
    float4 px[4], pbuf[4];

    // ---- prologue: prefetch + fuse chunk 0
#pragma unroll
    for (int j = 0; j < 4; ++j) {
        px[j] = *(const float4*)&x[(size_t)(bm + srow[j]) * D_DIM + sc4[j]];
        float4 wa = *(const float4*)&w [(size_t)(bn + srow[j]) * D_DIM + sc4[j]];
        float4 na = *(const float4*)&nw[(size_t)(bn + srow[j]) * D_DIM + sc4[j]];
        float4 fa = *(const float4*)&fm[sc4[j]];
        pbuf[j].x = wa.x + fa.x * na.x;
        pbuf[j].y = wa.y + fa.y * na.y;
        pbuf[j].z = wa.z + fa.z * na.z;
        pbuf[j].w = wa.w + fa.w * na.w;
    }

    for (int k0 = 0; k0 < D_DIM; k0 += BK) {
        // ---- commit staged registers to LDS
#pragma unroll
        for (int j = 0; j < 4; ++j) {
            *(float4*)&As[srow[j]][sc4[j]] = px[j];
            *(float4*)&Bs[srow[j]][sc4[j]] = pbuf[j];
        }
        __syncthreads();

        // ---- prefetch + fuse next chunk (overlaps the 64 WMMAs below)
        int kn = k0 + BK;
        if (kn < D_DIM) {
#pragma unroll
            for (int j = 0; j < 4; ++j) {
                px[j] = *(const float4*)&x[(size_t)(bm + srow[j]) * D_DIM + kn + sc4[j]];
                float4 wa = *(const float4*)&w [(size_t)(bn + srow[j]) * D_DIM + kn + sc4[j]];
                float4 na = *(const float4*)&nw[(size_t)(bn + srow[j]) * D_DIM + kn + sc4[j]];
                float4 fa = *(const float4*)&fm[kn + sc4[j]];
                pbuf[j].x = wa.x + fa.x * na.x;
                pbuf[j].y = wa.y + fa.y * na.y;
                pbuf[j].z = wa.z + fa.z * na.z;
                pbuf[j].w = wa.w + fa.w * na.w;
            }
        }

        // ---- 8 K-steps of 4, 8 WMMAs each (2x4 tile grid per wave)
#pragma unroll
        for (int kk = 0; kk < BK; kk += 4) {
            int k2 = kk + (hk << 1);   // lanes 0-15 -> kk,kk+1 ; lanes 16-31 -> kk+2,kk+3
            v2f af[2], bf[4];
#pragma unroll
            for (int tm = 0; tm < 2; ++tm) {
                int m = wm * 32 + tm * 16 + mrow;
                af[tm] = *(const v2f*)&As[m][k2];
            }
#pragma unroll
            for (int tn = 0; tn < 4; ++tn) {
                int n = wn * 64 + tn * 16 + mrow;
                bf[tn] = *(const v2f*)&Bs[n][k2];
            }
#pragma unroll
            for (int tm = 0; tm < 2; ++tm)
#pragma unroll
                for (int tn = 0; tn < 4; ++tn)
                    acc[tm][tn] = __builtin_amdgcn_wmma_f32_16x16x4_f32(
                        false, af[tm], false, bf[tn],
                        (short)0, acc[tm][tn], false, false);
        }
        __syncthreads();
    }

    // ---- store: C/D layout -> VGPR r holds M=r (lanes 0-15) / M=r+8 (lanes 16-31)
#pragma unroll
    for (int tm = 0; tm < 2; ++tm) {
#pragma unroll
        for (int tn = 0; tn < 4; ++tn) {
            int nloc = bn + wn * 64 + tn * 16 + mrow;
#pragma unroll
            for (int r = 0; r < 8; ++r) {
                int mloc = bm + wm * 32 + tm * 16 + r + (hk << 3);
                out[(size_t)mloc * O_DIM + nloc] = acc[tm][tn][r];
            }
        }
    }
}

// ---------------- Host-side launch ----------------
extern "C" void kernel_launch(void* const* d_in, const int* in_sizes, int n_in,
                              void* d_out, int out_size, void* d_ws, size_t ws_size,
                              hipStream_t stream) {
    const float*         x      = (const float*)d_in[0];
    const float*         w      = (const float*)d_in[1];
    const float*         nw     = (const float*)d_in[2];
    const int*           perm   = (const int*)d_in[3];
    const unsigned char* stored = (const unsigned char*)d_in[4];   // bool masks
    const int*           pb     = (const int*)d_in[5];

    char*     ws    = (char*)d_ws;
    float*    f     = (float*)(ws + 0);             // 4096 floats
    unsigned* mbits = (unsigned*)(ws + 16 * 1024);  // 128 u32
    int*      ovcnt = (int*)(ws + 17 * 1024);       // 100 ints
    float*    fmask = (float*)(ws + 20 * 1024);     // 4096 floats
    float*    out   = (float*)d_out;

    k_prompt_mean<<<D_DIM / 256, 256, 0, stream>>>(x, pb, f);
    k_topk_mask  <<<1,           256, 0, stream>>>(f, perm, mbits);
    k_overlap    <<<NMASKS,      256, 0, stream>>>(mbits, stored, ovcnt);
    k_select     <<<1,           256, 0, stream>>>(ovcnt, stored, fmask);

    dim3 grid(O_DIM / BN, M_DIM / BM);   // 32 x 64 workgroups
    k_gemm_wmma<<<grid, 256, 0, stream>>>(x, w, nw, fmask, out);
}